// BandSplitLinear_79139067396474
// MI455X (gfx1250) — compile-verified
//
#include <hip/hip_runtime.h>

// ---------------- band tables (compile-time constants from the reference) ----
__constant__ int BAND_START[45] = {
    0,4,9,15,22,30,39,49,60,72,85,99,114,130,147,165,184,204,225,247,
    270,294,319,345,372,400,429,459,490,522,554,586,618,650,682,714,
    746,778,810,842,874,906,938,970,1002};
__constant__ int BAND_LEN[45] = {
    4,5,6,7,8,9,10,11,12,13,14,15,16,17,18,19,20,21,22,23,
    24,25,26,27,28,29,30,31,32,32,32,32,32,32,32,32,
    32,32,32,32,32,32,32,32,23};

typedef __attribute__((ext_vector_type(2))) float v2f;
typedef __attribute__((ext_vector_type(8))) float v8f;

#define NK   45
#define INF  128    // fused in/out features (W*C)
#define DD   384
#define BT   8000   // B*T rows
#define TM   64     // rows per block in main kernel
#define LDA  132    // padded LDS row stride for A tile (conflict-free b64 A frags)
#define LDBI 288    // padded LDS row stride (floats) per K-pair of interleaved Wc

// Interleaved fused-weight global layout:
//   Wc_int[k][(i>>1)*256 + o*2 + (i&1)]  == Wc[k][i][o]
// so a lane's B fragment {B[k][n], B[k+1][n]} is one contiguous b64.

// ---------------- kernel 1: fuse Wc[k] = w_pre[k] (128x384) @ w_post[k] (384x128)
__global__ __launch_bounds__(256) void fuse_w_kernel(
    const float* __restrict__ w_pre, const float* __restrict__ w_post,
    float* __restrict__ WcI)
{
  const int k    = blockIdx.x;
  const int wave = threadIdx.x >> 5;
  const int lane = threadIdx.x & 31;
  const int lm   = lane & 15;   // M/N index within tile
  const int kh   = lane >> 4;   // K half select

  const float* wp = w_pre  + (size_t)k * INF * DD;   // [i][d] row-major
  const float* wq = w_post + (size_t)k * DD * INF;   // [d][o] row-major
  float* dst = WcI + (size_t)k * INF * INF;          // interleaved layout

  for (int tix = wave; tix < 64; tix += 8) {         // 8x8 grid of 16x16 tiles
    const int mi = tix >> 3;
    const int no = tix & 7;
    v8f acc = {};
    const float* ap = wp + (size_t)(mi * 16 + lm) * DD;
    const float* bp = wq + no * 16 + lm;
    for (int d0 = 0; d0 < DD; d0 += 4) {
      v2f a = *(const v2f*)&ap[d0 + 2 * kh];
      v2f b;
      b.x = bp[(size_t)(d0 + 2 * kh) * INF];
      b.y = bp[(size_t)(d0 + 2 * kh + 1) * INF];
      acc = __builtin_amdgcn_wmma_f32_16x16x4_f32(
          false, a, false, b, (short)0, acc, false, false);
    }
    // D frag: VGPR r -> M(i) = r + 8*kh, N(o) = lm ; store interleaved
#pragma unroll
    for (int r = 0; r < 8; ++r) {
      const int i = mi * 16 + r + kh * 8;
      const int o = no * 16 + lm;
      dst[(size_t)(i >> 1) * 256 + o * 2 + (i & 1)] = acc[r];
    }
  }
}

// ---------------- kernel 2: bc[k] = b_pre[k] @ w_post[k] + b_post[k] ----------
__global__ __launch_bounds__(128) void fuse_b_kernel(
    const float* __restrict__ b_pre, const float* __restrict__ w_post,
    const float* __restrict__ b_post, float* __restrict__ bc)
{
  const int k = blockIdx.x;
  const int o = threadIdx.x; // 0..127
  const float* wq = w_post + (size_t)k * DD * INF;
  float acc = b_post[k * INF + o];
  for (int d = 0; d < DD; ++d)
    acc += b_pre[k * DD + d] * wq[(size_t)d * INF + o];
  bc[k * INF + o] = acc;
}

// ---------------- kernel 3: main fused band-split linear ---------------------
// grid = (125 m-tiles of 64 rows, 45 bands), block = 256 (8 waves)
// wave -> 2x2 register block of 16x16 tiles: 2 A-frag + 2 B-frag b64 loads
// feed 4 WMMAs per K-step (1.0 DS op / WMMA, all conflict-free b64).
__global__ __launch_bounds__(256) void bsl_main_kernel(
    const float* __restrict__ x,    // (B=8, C=4, T=1000, F=1025)
    const float* __restrict__ WcI,  // (45, 64, 256) interleaved fused weights
    const float* __restrict__ bc,   // (45, 128)
    float* __restrict__ out)        // (B, C, T, F)
{
  extern __shared__ float smem[];
  float* sA    = smem;                  // [TM][LDA]
  float* sB    = smem + TM * LDA;       // [64 K-pairs][LDBI]
  float* sBias = sB + 64 * LDBI;        // [128]

  const int k     = blockIdx.y;
  const int mbase = blockIdx.x * TM;    // row base in [0, 8000)
  const int tid   = threadIdx.x;
  const int start = BAND_START[k];
  const int len   = BAND_LEN[k];

  // stage interleaved fused weights (64 KB) into LDS, contiguous float4 copies
  {
    const float4* wck = (const float4*)(WcI + (size_t)k * INF * INF);
    for (int e = tid; e < INF * INF / 4; e += 256) {
      const int kp = (e * 4) >> 8;     // K-pair row (0..63)
      const int c0 = (e * 4) & 255;    // column within interleaved row
      const float4 v = wck[e];
      float* d = &sB[kp * LDBI + c0];
      d[0] = v.x; d[1] = v.y; d[2] = v.z; d[3] = v.w;
    }
  }
  if (tid < INF) sBias[tid] = bc[k * INF + tid];

  // gather A tile: 64 rows x 128 features (feature i = w*4 + c)
  for (int e = tid; e < TM * INF; e += 256) {
    const int r = e >> 7;
    const int i = e & 127;
    const int w = i >> 2, c = i & 3;
    float v = 0.0f;
    if (w < len) {
      const int row = mbase + r;
      const int b = row / 1000;
      const int t = row - b * 1000;
      v = x[(((size_t)(b * 4 + c)) * 1000 + t) * 1025 + (start + w)];
    }
    sA[r * LDA + i] = v;
  }
  __syncthreads();

  // 8 waves -> (2 m-tiles) x (2 n-tiles) each over a 4x8 tile grid
  const int wave = tid >> 5;
  const int lane = tid & 31;
  const int mp   = wave >> 2;          // m-pair 0..1 -> m-tiles {2mp, 2mp+1}
  const int np   = wave & 3;           // n-pair 0..3 -> n-tiles {2np, 2np+1}
  const int lm   = lane & 15;
  const int kh   = lane >> 4;

  v8f acc00 = {}, acc01 = {}, acc10 = {}, acc11 = {};
  const float* aRow0 = &sA[((2 * mp) * 16 + lm) * LDA];
  const float* aRow1 = &sA[((2 * mp + 1) * 16 + lm) * LDA];
  const float* bRow0 = &sB[kh * LDBI + (2 * np) * 16 * 2 + lm * 2];
  const float* bRow1 = &sB[kh * LDBI + (2 * np + 1) * 16 * 2 + lm * 2];

#pragma unroll 4
  for (int k0 = 0; k0 < INF; k0 += 4) {
    v2f a0 = *(const v2f*)&aRow0[k0 + 2 * kh];
    v2f a1 = *(const v2f*)&aRow1[k0 + 2 * kh];
    v2f b0 = *(const v2f*)&bRow0[(k0 >> 1) * LDBI];
    v2f b1 = *(const v2f*)&bRow1[(k0 >> 1) * LDBI];
    acc00 = __builtin_amdgcn_wmma_f32_16x16x4_f32(
        false, a0, false, b0, (short)0, acc00, false, false);
    acc01 = __builtin_amdgcn_wmma_f32_16x16x4_f32(
        false, a0, false, b1, (short)0, acc01, false, false);
    acc10 = __builtin_amdgcn_wmma_f32_16x16x4_f32(
        false, a1, false, b0, (short)0, acc10, false, false);
    acc11 = __builtin_amdgcn_wmma_f32_16x16x4_f32(
        false, a1, false, b1, (short)0, acc11, false, false);
  }

  const int o0 = (2 * np) * 16 + lm;
  const int o1 = (2 * np + 1) * 16 + lm;
  const float bias0 = sBias[o0];
  const float bias1 = sBias[o1];
  const int w0 = o0 >> 2, c0 = o0 & 3;
  const int w1 = o1 >> 2, c1 = o1 & 3;

  // scatter store: bands are disjoint and cover all bins -> plain stores
#pragma unroll
  for (int r = 0; r < 8; ++r) {
    const int mloc = r + kh * 8;       // D frag: M = r + 8*(lane/16)
    {
      const int row = mbase + (2 * mp) * 16 + mloc;
      const int b = row / 1000;
      const int t = row - b * 1000;
      const size_t base = (((size_t)b * 4) * 1000 + t) * 1025 + (start);
      if (w0 < len)
        out[base + (size_t)c0 * 1000 * 1025 + w0] = acc00[r] + bias0;
      if (w1 < len)
        out[base + (size_t)c1 * 1000 * 1025 + w1] = acc01[r] + bias1;
    }
    {
      const int row = mbase + (2 * mp + 1) * 16 + mloc;
      const int b = row / 1000;
      const int t = row - b * 1000;
      const size_t base = (((size_t)b * 4) * 1000 + t) * 1025 + (start);
      if (w0 < len)
        out[base + (size_t)c0 * 1000 * 1025 + w0] = acc10[r] + bias0;
      if (w1 < len)
        out[base + (size_t)c1 * 1000 * 1025 + w1] = acc11[r] + bias1;
    }
  }
}

// ---------------- launch ------------------------------------------------------
extern "C" void kernel_launch(void* const* d_in, const int* in_sizes, int n_in,
                              void* d_out, int out_size, void* d_ws, size_t ws_size,
                              hipStream_t stream)
{
  const float* x      = (const float*)d_in[0];
  const float* w_pre  = (const float*)d_in[1];
  const float* b_pre  = (const float*)d_in[2];
  const float* w_post = (const float*)d_in[3];
  const float* b_post = (const float*)d_in[4];
  float* out = (float*)d_out;

  float* WcI = (float*)d_ws;                   // 45*128*128 f32 = 2.95 MB
  float* bc  = WcI + (size_t)NK * INF * INF;   // 45*128 f32

  fuse_w_kernel<<<NK, 256, 0, stream>>>(w_pre, w_post, WcI);
  fuse_b_kernel<<<NK, 128, 0, stream>>>(b_pre, w_post, b_post, bc);

  const size_t smem = (size_t)(TM * LDA + 64 * LDBI + INF) * sizeof(float);
  bsl_main_kernel<<<dim3(BT / TM, NK), 256, smem, stream>>>(x, WcI, bc, out);
}